// HungarianMatcher_55499567399183
// MI455X (gfx1250) — compile-verified
//
#include <hip/hip_runtime.h>
#include <math.h>

// Problem constants (from setup_inputs: B=16, Q=900, C=92, n_per=32)
#define B_   16
#define Q_   900
#define C_   92
#define NPER 32
#define TTOT 512            // B_ * NPER
#define SOLVER_THREADS 256

typedef __attribute__((ext_vector_type(2))) float v2f;
typedef __attribute__((ext_vector_type(8))) float v8f;

// ---------------------------------------------------------------------------
// Kernel 1: row-wise softmax of logits -> probs.  One wave32 per row.
// ---------------------------------------------------------------------------
__global__ void hm_softmax_rows(const float* __restrict__ logits,
                                float* __restrict__ probs) {
    const int row  = blockIdx.x;          // 0 .. B_*Q_-1
    const int lane = threadIdx.x;         // 0 .. 31
    const float* lg = logits + (size_t)row * C_;
    float m = -3.4e38f;
    for (int c = lane; c < C_; c += 32) m = fmaxf(m, lg[c]);
    for (int off = 16; off > 0; off >>= 1) m = fmaxf(m, __shfl_xor(m, off));
    float s = 0.f;
    for (int c = lane; c < C_; c += 32) s += expf(lg[c] - m);
    for (int off = 16; off > 0; off >>= 1) s += __shfl_xor(s, off);
    const float inv = 1.0f / s;
    float* pr = probs + (size_t)row * C_;
    for (int c = lane; c < C_; c += 32) pr[c] = expf(lg[c] - m) * inv;
}

// ---------------------------------------------------------------------------
// Kernel 2: class cost via f32 WMMA.  D = P(16x92 tile) x onehot(92x16) is an
// EXACT gather of softmax probs at the target labels (one-hot matmul, fp32).
// Writes costT[t][q] = COST_CLASS * (-prob[q, label_t]).  One wave per tile.
// A layout (16x4 f32): lane<16 -> M=lane, K={0,1}; lane>=16 -> M=lane-16, K={2,3}
// B layout (4x16 f32): VGPR j, lane-half h -> K=j+2h, N=lane&15
// D layout (16x16 f32): VGPR r, half h -> M=r+8h, N=lane&15
//
// Out-of-range query rows (q >= 900, only in the last M tile) are CLAMPED to
// row 0 instead of masked: the bogus products only land in D rows that the
// store guard discards, so A loads stay unconditional (full-EXEC, b64 loads,
// no exec save/restore around the WMMA stream).
// ---------------------------------------------------------------------------
__global__ void hm_class_cost_wmma(const float* __restrict__ probs,
                                   const int*  __restrict__ labels,
                                   float* __restrict__ costT) {
    const int mtile = blockIdx.x;   // 0..56 (57 tiles of 16 queries, last padded)
    const int ntile = blockIdx.y;   // 0..1  (32 targets per batch)
    const int b     = blockIdx.z;   // 0..15
    const int lane  = threadIdx.x;  // 0..31
    const int half  = lane >> 4;
    const int l     = lane & 15;

    const int q  = mtile * 16 + l;
    const int qc = (q < Q_) ? q : 0;             // clamp; rows >=Q_ discarded at store
    // Row base is 92 floats = 368 bytes (8B aligned); k0 is even -> aligned v2f.
    const v2f* Prow = (const v2f*)(probs + (size_t)(b * Q_ + qc) * C_);

    const int t   = b * NPER + ntile * 16 + l;   // B-matrix column N = l
    const int lab = labels[t];

    v8f acc = {};
#pragma unroll
    for (int kk = 0; kk < C_; kk += 4) {         // C_ = 92 = 23 * 4
        const int k0 = kk + 2 * half;            // even
        const v2f a = Prow[k0 >> 1];             // {P[q][k0], P[q][k0+1]}
        v2f bb;
        bb.x = (lab == k0)     ? 1.f : 0.f;
        bb.y = (lab == k0 + 1) ? 1.f : 0.f;
        acc = __builtin_amdgcn_wmma_f32_16x16x4_f32(
            /*neg_a=*/false, a, /*neg_b=*/false, bb,
            /*c_mod=*/(short)0, acc, /*reuse_a=*/false, /*reuse_b=*/false);
    }

    const int qbase = mtile * 16 + 8 * half;     // D rows M = r + 8*half
    float* dst = costT + (size_t)t * Q_ + qbase;
#pragma unroll
    for (int r = 0; r < 8; ++r) {
        if (qbase + r < Q_) dst[r] = -acc[r];    // COST_CLASS = 1.0
    }
}

// ---------------------------------------------------------------------------
// Kernel 3: add 5*L1 + 2*(-GIoU) into costT[t][q].  Coalesced over q.
// ---------------------------------------------------------------------------
__global__ void hm_box_cost(const float* __restrict__ pred_boxes,
                            const float* __restrict__ tgt_boxes,
                            float* __restrict__ costT) {
    const int idx = blockIdx.x * blockDim.x + threadIdx.x;
    if (idx >= TTOT * Q_) return;
    const int t = idx / Q_;
    const int q = idx - t * Q_;
    const int b = t >> 5;                 // NPER = 32

    const float* pb = pred_boxes + (size_t)(b * Q_ + q) * 4;
    const float* tb = tgt_boxes + (size_t)t * 4;

    const float l1 = fabsf(pb[0] - tb[0]) + fabsf(pb[1] - tb[1]) +
                     fabsf(pb[2] - tb[2]) + fabsf(pb[3] - tb[3]);

    // cxcywh -> xyxy
    const float px1 = pb[0] - 0.5f * pb[2], py1 = pb[1] - 0.5f * pb[3];
    const float px2 = pb[0] + 0.5f * pb[2], py2 = pb[1] + 0.5f * pb[3];
    const float tx1 = tb[0] - 0.5f * tb[2], ty1 = tb[1] - 0.5f * tb[3];
    const float tx2 = tb[0] + 0.5f * tb[2], ty2 = tb[1] + 0.5f * tb[3];

    const float area1 = (px2 - px1) * (py2 - py1);
    const float area2 = (tx2 - tx1) * (ty2 - ty1);
    const float iw = fmaxf(fminf(px2, tx2) - fmaxf(px1, tx1), 0.f);
    const float ih = fmaxf(fminf(py2, ty2) - fmaxf(py1, ty1), 0.f);
    const float inter = iw * ih;
    const float uni   = area1 + area2 - inter;
    const float iou   = inter / uni;
    const float cw = fmaxf(fmaxf(px2, tx2) - fminf(px1, tx1), 0.f);
    const float ch = fmaxf(fmaxf(py2, ty2) - fminf(py1, ty1), 0.f);
    const float areac = cw * ch;
    const float giou  = iou - (areac - uni) / areac;

    costT[idx] += 5.0f * l1 - 2.0f * giou;   // COST_BBOX=5, COST_GIOU=2
}

// ---------------------------------------------------------------------------
// Kernel 4: Jonker-Volgenant LSA per batch (transposed: n=32 rows (targets),
// m=900 cols (queries)), float64 duals to mirror the reference numerics.
// One workgroup per batch; inner O(m) passes parallelized across threads;
// argmin reduction reproduces np.argmin first-minimum tie-breaking.
// ---------------------------------------------------------------------------
__global__ void hm_jv_solve(const float* __restrict__ costT,
                            float* __restrict__ sol) {
    const int b   = blockIdx.x;
    const int tid = threadIdx.x;
    const int nthr = SOLVER_THREADS;
    const int n = NPER;      // 32 rows
    const int m = Q_;        // 900 cols
    const double INF = 1e18;

    __shared__ double sh_v[Q_ + 1];
    __shared__ double sh_minv[Q_ + 1];
    __shared__ double sh_u[NPER + 1];
    __shared__ int    sh_p[Q_ + 1];
    __shared__ int    sh_way[Q_ + 1];
    __shared__ unsigned char sh_used[Q_ + 1];
    __shared__ int    sh_usedList[NPER + 4];
    __shared__ int    sh_usedCnt;
    __shared__ double red_val[SOLVER_THREADS];
    __shared__ int    red_idx[SOLVER_THREADS];

    const float* Cb = costT + (size_t)b * NPER * Q_;   // Cb[i*Q_ + j] = c[i][j]

    for (int j = tid; j <= m; j += nthr) { sh_v[j] = 0.0; sh_p[j] = 0; sh_way[j] = 0; }
    for (int i = tid; i <= n; i += nthr) sh_u[i] = 0.0;
    __syncthreads();

    for (int i = 1; i <= n; ++i) {
        if (tid == 0) { sh_p[0] = i; sh_usedCnt = 0; }
        for (int j = tid; j <= m; j += nthr) { sh_minv[j] = INF; sh_used[j] = 0; }
        __syncthreads();

        int j0 = 0;
        while (true) {
            if (tid == 0) { sh_used[j0] = 1; sh_usedList[sh_usedCnt++] = j0; }
            __syncthreads();

            const int i0 = sh_p[j0];
            const double ui0 = sh_u[i0];
            const float* crow = Cb + (size_t)(i0 - 1) * Q_;

            double lmin = INF;
            int    lj   = m + 1;
            for (int j = tid + 1; j <= m; j += nthr) {
                if (!sh_used[j]) {
                    const double cur = (double)crow[j - 1] - ui0 - sh_v[j];
                    if (cur < sh_minv[j]) { sh_minv[j] = cur; sh_way[j] = j0; }
                    const double mv = sh_minv[j];
                    if (mv < lmin) { lmin = mv; lj = j; }  // strict < keeps first
                }
            }
            red_val[tid] = lmin;
            red_idx[tid] = lj;
            __syncthreads();
            for (int s = nthr >> 1; s > 0; s >>= 1) {
                if (tid < s) {
                    const double ov = red_val[tid + s];
                    const int    oi = red_idx[tid + s];
                    if (ov < red_val[tid] ||
                        (ov == red_val[tid] && oi < red_idx[tid])) {
                        red_val[tid] = ov; red_idx[tid] = oi;
                    }
                }
                __syncthreads();
            }
            const double delta = red_val[0];
            const int    j1    = red_idx[0];

            if (tid == 0) {
                for (int k = 0; k < sh_usedCnt; ++k) {
                    const int jj = sh_usedList[k];
                    sh_u[sh_p[jj]] += delta;
                    sh_v[jj]       -= delta;
                }
            }
            __syncthreads();
            for (int j = tid + 1; j <= m; j += nthr)
                if (!sh_used[j]) sh_minv[j] -= delta;
            __syncthreads();

            j0 = j1;
            if (sh_p[j0] == 0) break;
        }

        if (tid == 0) {            // augmenting path (<= n+1 steps)
            int jj = j0;
            while (jj) { const int jp = sh_way[jj]; sh_p[jj] = sh_p[jp]; jj = jp; }
        }
        __syncthreads();
    }

    // sol[q][b*32 + (target)] = 1 for assigned (query=j-1, target=p[j]-1)
    for (int j = tid + 1; j <= m; j += nthr) {
        const int i = sh_p[j];
        if (i > 0) sol[(size_t)(j - 1) * TTOT + b * NPER + (i - 1)] = 1.0f;
    }
}

// ---------------------------------------------------------------------------
extern "C" void kernel_launch(void* const* d_in, const int* in_sizes, int n_in,
                              void* d_out, int out_size, void* d_ws, size_t ws_size,
                              hipStream_t stream) {
    (void)in_sizes; (void)n_in; (void)out_size; (void)ws_size;
    const float* pred_logits = (const float*)d_in[0];  // [16,900,92]
    const float* pred_boxes  = (const float*)d_in[1];  // [16,900,4]
    const int*   tgt_labels  = (const int*)d_in[2];    // [512]
    const float* tgt_boxes   = (const float*)d_in[3];  // [512,4]
    // d_in[4] = n_targets (==32), compile-time constant NPER

    float* out   = (float*)d_out;                      // [900,512]
    float* costT = (float*)d_ws;                       // [512,900]
    float* probs = costT + (size_t)TTOT * Q_;          // [16*900,92]

    hipMemsetAsync(out, 0, (size_t)Q_ * TTOT * sizeof(float), stream);

    hm_softmax_rows<<<B_ * Q_, 32, 0, stream>>>(pred_logits, probs);

    dim3 gcls((Q_ + 15) / 16, NPER / 16, B_);          // 57 x 2 x 16
    hm_class_cost_wmma<<<gcls, 32, 0, stream>>>(probs, tgt_labels, costT);

    const int nel = TTOT * Q_;
    hm_box_cost<<<(nel + 255) / 256, 256, 0, stream>>>(pred_boxes, tgt_boxes, costT);

    hm_jv_solve<<<B_, SOLVER_THREADS, 0, stream>>>(costT, out);
}